// WeightedGIN_46626164965920
// MI455X (gfx1250) — compile-verified
//
#include <hip/hip_runtime.h>
#include <hip/hip_bf16.h>

#define N_NODES 40000
#define N_EDGES 640000
#define DIM     128
#define NGRAPH  64
#define LDS_PAD 132   // 16 A-rows padded: bank = (132*row + col) % 64 = 4*row + col -> conflict-free b64 reads

typedef __attribute__((ext_vector_type(2))) float v2f;
typedef __attribute__((ext_vector_type(8))) float v8f;

// ---------- wave32 all-lanes sum ----------
__device__ __forceinline__ float wave_sum(float v) {
    v += __shfl_xor(v, 16, 32);
    v += __shfl_xor(v, 8, 32);
    v += __shfl_xor(v, 4, 32);
    v += __shfl_xor(v, 2, 32);
    v += __shfl_xor(v, 1, 32);
    return v;
}

// ---------- embedding gather: x[i,:] = emb[node_ids[i],:] ----------
__global__ __launch_bounds__(256) void k_gather(const int* __restrict__ ids,
                                                const float* __restrict__ emb,
                                                float* __restrict__ x) {
    int t = blockIdx.x * blockDim.x + threadIdx.x;   // one float4 per thread
    int node = t >> 5;
    int lane = t & 31;
    if (node >= N_NODES) return;
    const float4 v = *(const float4*)(emb + (size_t)ids[node] * DIM + lane * 4);
    *(float4*)(x + (size_t)node * DIM + lane * 4) = v;
}

// ---------- LayerNorm, fused with self-loop term hpre = (2+eps)*xn ----------
__global__ __launch_bounds__(256) void k_ln(const float* __restrict__ x,
                                            const float* __restrict__ g,
                                            const float* __restrict__ b,
                                            const float* __restrict__ epsp,
                                            float* __restrict__ xn,
                                            float* __restrict__ hpre) {
    int node = blockIdx.x * (blockDim.x >> 5) + (threadIdx.x >> 5);
    int lane = threadIdx.x & 31;
    if (node >= N_NODES) return;
    const float4 v = *(const float4*)(x + (size_t)node * DIM + lane * 4);
    float s = v.x + v.y + v.z + v.w;
    s = wave_sum(s);
    float mu = s * (1.0f / DIM);
    float dx = v.x - mu, dy = v.y - mu, dz = v.z - mu, dw = v.w - mu;
    float q = dx * dx + dy * dy + dz * dz + dw * dw;
    q = wave_sum(q);
    float inv = __frsqrt_rn(q * (1.0f / DIM) + 1e-5f);
    const float4 gg = *(const float4*)(g + lane * 4);
    const float4 bb = *(const float4*)(b + lane * 4);
    float4 o;
    o.x = dx * inv * gg.x + bb.x;
    o.y = dy * inv * gg.y + bb.y;
    o.z = dz * inv * gg.z + bb.z;
    o.w = dw * inv * gg.w + bb.w;
    *(float4*)(xn + (size_t)node * DIM + lane * 4) = o;
    float c = 2.0f + epsp[0];
    float4 h;
    h.x = c * o.x; h.y = c * o.y; h.z = c * o.z; h.w = c * o.w;
    *(float4*)(hpre + (size_t)node * DIM + lane * 4) = h;
}

// ---------- edge scatter: hpre[dst,:] += w_e * xn[src,:] (L2 f32 atomics) ----------
__global__ __launch_bounds__(256) void k_scatter(const int* __restrict__ src,
                                                 const int* __restrict__ dst,
                                                 const float* __restrict__ ew,
                                                 const float* __restrict__ xn,
                                                 float* __restrict__ hpre) {
    int e = blockIdx.x * (blockDim.x >> 5) + (threadIdx.x >> 5);
    if (e >= N_EDGES) return;
    int lane = threadIdx.x & 31;
    int s = src[e], d = dst[e];
    float w = ew[e];
    const float4 v = *(const float4*)(xn + (size_t)s * DIM + lane * 4);
    float* o = hpre + (size_t)d * DIM + lane * 4;
    atomicAdd(o + 0, w * v.x);
    atomicAdd(o + 1, w * v.y);
    atomicAdd(o + 2, w * v.z);
    atomicAdd(o + 3, w * v.w);
}

// ---------- weight repack: Wp[k/2][n][k%2] = W[k][n] (makes B fragments one b64 load) ----------
__global__ __launch_bounds__(256) void k_pack(const float* __restrict__ W,
                                              float* __restrict__ Wp,
                                              int K, int Ncols) {
    int i = blockIdx.x * blockDim.x + threadIdx.x;
    if (i >= K * Ncols) return;
    int k = i / Ncols, n = i - k * Ncols;
    Wp[(size_t)(k >> 1) * Ncols * 2 + n * 2 + (k & 1)] = W[i];
}

// ---------- fp32 WMMA GEMM: C[M,Ncols] = act(A[M,K] @ W[K,Ncols] + bias) ----------
// Block = (Ncols/16) waves; stages the shared 16xK A tile in LDS (conflict-free
// padded layout), weights come prepacked so each B fragment is one b64 load.
// Inner loop: ds_load_b64 (A) + global_load_b64 (B) + v_wmma_f32_16x16x4_f32.
template <bool RELU>
__global__ __launch_bounds__(256) void k_gemm(const float* __restrict__ A,
                                              const float* __restrict__ Wp,
                                              const float* __restrict__ bias,
                                              float* __restrict__ C,
                                              int K, int Ncols) {
    __shared__ float As[16 * LDS_PAD];
    const int tid = threadIdx.x;
    const int rowBase = blockIdx.x * 16;

    // stage A tile: 16 rows x K cols, float4 per thread-iteration
    for (int i = tid; i < 16 * (K / 4); i += blockDim.x) {
        int r  = i / (K / 4);
        int c4 = i - r * (K / 4);
        const float4 v = *(const float4*)(A + (size_t)(rowBase + r) * K + c4 * 4);
        *(float4*)(&As[r * LDS_PAD + c4 * 4]) = v;
    }
    __syncthreads();

    const int wave = tid >> 5;
    const int lane = tid & 31;
    const int half = lane >> 4;       // 0: K pair {0,1}; 1: K pair {2,3}
    const int lm   = lane & 15;       // row (A) / col (B,C) within tile
    const int colBase = wave * 16;

    v8f acc = {};
    const float* arow = &As[lm * LDS_PAD + 2 * half];
    const float* bptr = Wp + (size_t)half * Ncols * 2 + (colBase + lm) * 2;

#pragma unroll 4
    for (int k0 = 0; k0 < K; k0 += 4) {
        const float2 av = *(const float2*)(arow + k0);                       // LDS b64
        const float2 bv = *(const float2*)(bptr + (size_t)(k0 >> 1) * Ncols * 2); // global b64
        v2f a; a.x = av.x; a.y = av.y;
        v2f b; b.x = bv.x; b.y = bv.y;
        acc = __builtin_amdgcn_wmma_f32_16x16x4_f32(
            /*neg_a=*/false, a, /*neg_b=*/false, b,
            /*c_mod=*/(short)0, acc, /*reuse_a=*/false, /*reuse_b=*/false);
    }

    const float bvv = bias[colBase + lm];
#pragma unroll
    for (int r = 0; r < 8; ++r) {
        int row = rowBase + r + 8 * half;                  // C layout: VGPR r -> M=r / M=r+8
        float val = acc[r] + bvv;
        if (RELU) val = fmaxf(val, 0.0f);
        C[(size_t)row * Ncols + colBase + lm] = val;
    }
}

// ---------- zero scratch ----------
__global__ __launch_bounds__(256) void k_zero(float* __restrict__ p, int n) {
    int i = blockIdx.x * blockDim.x + threadIdx.x;
    if (i < n) p[i] = 0.0f;
}

// ---------- mean-pool accumulate ----------
__global__ __launch_bounds__(256) void k_pool(const int* __restrict__ batch,
                                              const float* __restrict__ x,
                                              float* __restrict__ sums,
                                              float* __restrict__ cnt) {
    int node = blockIdx.x * (blockDim.x >> 5) + (threadIdx.x >> 5);
    if (node >= N_NODES) return;
    int lane = threadIdx.x & 31;
    int g = batch[node];
    const float4 v = *(const float4*)(x + (size_t)node * DIM + lane * 4);
    float* o = sums + (size_t)g * DIM + lane * 4;
    atomicAdd(o + 0, v.x);
    atomicAdd(o + 1, v.y);
    atomicAdd(o + 2, v.z);
    atomicAdd(o + 3, v.w);
    if (lane == 0) atomicAdd(cnt + g, 1.0f);
}

__global__ __launch_bounds__(256) void k_pooldiv(const float* __restrict__ sums,
                                                 const float* __restrict__ cnt,
                                                 float* __restrict__ g) {
    int i = blockIdx.x * blockDim.x + threadIdx.x;
    if (i < NGRAPH * DIM) {
        float c = cnt[i / DIM];
        g[i] = sums[i] / fmaxf(c, 1.0f);
    }
}

// ---------- tiny head: out[64,2] = hid[64,64] @ W4[64,2] + b4 ----------
__global__ __launch_bounds__(128) void k_head2(const float* __restrict__ hid,
                                               const float* __restrict__ W4,
                                               const float* __restrict__ b4,
                                               float* __restrict__ out) {
    int i = threadIdx.x;
    if (i >= NGRAPH * 2) return;
    int g = i >> 1, o = i & 1;
    float s = b4[o];
    for (int k = 0; k < 64; ++k) s += hid[g * 64 + k] * W4[k * 2 + o];
    out[i] = s;
}

extern "C" void kernel_launch(void* const* d_in, const int* in_sizes, int n_in,
                              void* d_out, int out_size, void* d_ws, size_t ws_size,
                              hipStream_t stream) {
    (void)in_sizes; (void)n_in; (void)out_size; (void)ws_size;

    // setup_inputs() depth-first flattening:
    const int*   node_ids = (const int*)d_in[0];
    const int*   src      = (const int*)d_in[1];       // edge_index row 0
    const int*   dst      = src + N_EDGES;             // edge_index row 1
    const float* ew       = (const float*)d_in[2];
    const int*   batch    = (const int*)d_in[3];
    const float* emb      = (const float*)d_in[4];
    const float* l_lng[2] = {(const float*)d_in[5],  (const float*)d_in[12]};
    const float* l_lnb[2] = {(const float*)d_in[6],  (const float*)d_in[13]};
    const float* l_eps[2] = {(const float*)d_in[7],  (const float*)d_in[14]};
    const float* l_W1[2]  = {(const float*)d_in[8],  (const float*)d_in[15]};
    const float* l_b1[2]  = {(const float*)d_in[9],  (const float*)d_in[16]};
    const float* l_W2[2]  = {(const float*)d_in[10], (const float*)d_in[17]};
    const float* l_b2[2]  = {(const float*)d_in[11], (const float*)d_in[18]};
    const float* W3       = (const float*)d_in[19];
    const float* b3       = (const float*)d_in[20];
    const float* W4       = (const float*)d_in[21];
    const float* b4       = (const float*)d_in[22];
    float* out = (float*)d_out;

    // workspace layout
    float* ws   = (float*)d_ws;
    float* x    = ws;                                   // N*DIM
    float* xn   = x    + (size_t)N_NODES * DIM;         // N*DIM
    float* hpre = xn   + (size_t)N_NODES * DIM;         // N*DIM
    float* t    = hpre + (size_t)N_NODES * DIM;         // N*DIM
    float* sums = t    + (size_t)N_NODES * DIM;         // NGRAPH*DIM
    float* cnt  = sums + NGRAPH * DIM;                  // NGRAPH (contiguous after sums)
    float* gp   = cnt  + NGRAPH;                        // NGRAPH*DIM
    float* hid  = gp   + NGRAPH * DIM;                  // NGRAPH*64
    float* W1p[2], *W2p[2];
    float* packs = hid + NGRAPH * 64;
    W1p[0] = packs;                 // DIM*DIM each
    W2p[0] = W1p[0] + DIM * DIM;
    W1p[1] = W2p[0] + DIM * DIM;
    W2p[1] = W1p[1] + DIM * DIM;
    float* W3p = W2p[1] + DIM * DIM;                    // DIM*64

    const int waveBlocksN = (N_NODES + 7) / 8;          // 8 waves (nodes) per 256-thr block
    const int waveBlocksE = (N_EDGES + 7) / 8;

    // repack all GEMM weights once per call (tiny)
    for (int l = 0; l < 2; ++l) {
        k_pack<<<(DIM * DIM + 255) / 256, 256, 0, stream>>>(l_W1[l], W1p[l], DIM, DIM);
        k_pack<<<(DIM * DIM + 255) / 256, 256, 0, stream>>>(l_W2[l], W2p[l], DIM, DIM);
    }
    k_pack<<<(DIM * 64 + 255) / 256, 256, 0, stream>>>(W3, W3p, DIM, 64);

    k_gather<<<(N_NODES * 32 + 255) / 256, 256, 0, stream>>>(node_ids, emb, x);

    for (int l = 0; l < 2; ++l) {
        k_ln<<<waveBlocksN, 256, 0, stream>>>(x, l_lng[l], l_lnb[l], l_eps[l], xn, hpre);
        k_scatter<<<waveBlocksE, 256, 0, stream>>>(src, dst, ew, xn, hpre);
        // h = relu(hpre @ W1 + b1); x = relu(h @ W2 + b2)
        k_gemm<true><<<N_NODES / 16, 256, 0, stream>>>(hpre, W1p[l], l_b1[l], t, DIM, DIM);
        k_gemm<true><<<N_NODES / 16, 256, 0, stream>>>(t, W2p[l], l_b2[l], x, DIM, DIM);
    }

    k_zero<<<(NGRAPH * DIM + NGRAPH + 255) / 256, 256, 0, stream>>>(sums, NGRAPH * DIM + NGRAPH);
    k_pool<<<waveBlocksN, 256, 0, stream>>>(batch, x, sums, cnt);
    k_pooldiv<<<(NGRAPH * DIM + 255) / 256, 256, 0, stream>>>(sums, cnt, gp);

    // head: hid = relu(gp @ W3 + b3)  (64x128 @ 128x64 -> 4 row tiles x 4 col tiles)
    k_gemm<true><<<NGRAPH / 16, 128, 0, stream>>>(gp, W3p, b3, hid, DIM, 64);
    k_head2<<<1, 128, 0, stream>>>(hid, W4, b4, out);
}